// ExplicitCircuit_57123065037204
// MI455X (gfx1250) — compile-verified
//
#include <hip/hip_runtime.h>
#include <math.h>

// 12-qubit batched state-vector simulator, one workgroup (256 thr) per batch element.
// State: 4096 complex amps; thread t holds amps [t*16, t*16+16) (qubits 8..11 local,
// qubits 3..7 = lane bits via shfl_xor, qubits 0..2 = wave bits via LDS exchange).
// Layer local gates (qubits 8..11) fused into a 16x16 complex matrix applied with
// V_WMMA_F32_16X16X4_F32. Inputs staged into LDS with global_load_async_to_lds_b32.

#define NQ     12
#define BLOCK  256
#define LOCAL  16
#define IN_DIM 78
#define NPAIR  66
#define NWTS   108

typedef float v2f __attribute__((ext_vector_type(2)));
typedef float v8f __attribute__((ext_vector_type(8)));

// u[8] = {u00r,u00i, u01r,u01i, u10r,u10i, u11r,u11i} of U = RZ(w2)*RY(w1)*RX(w0)
__device__ __forceinline__ void fused_u(float w0, float w1, float w2, float u[8]) {
  float c0, s0, c1, s1, cz, sz;
  sincosf(0.5f * w0, &s0, &c0);
  sincosf(0.5f * w1, &s1, &c1);
  sincosf(0.5f * w2, &sz, &cz);
  // M = RY * RX ; RX = [[c0,-i s0],[-i s0,c0]], RY = [[c1,-s1],[s1,c1]]
  const float m00r =  c1 * c0, m00i =  s1 * s0;
  const float m01r = -s1 * c0, m01i = -c1 * s0;
  const float m10r =  s1 * c0, m10i = -c1 * s0;
  const float m11r =  c1 * c0, m11i = -s1 * s0;
  // row0 *= (cz - i sz), row1 *= (cz + i sz)
  u[0] = cz * m00r + sz * m00i; u[1] = cz * m00i - sz * m00r;
  u[2] = cz * m01r + sz * m01i; u[3] = cz * m01i - sz * m01r;
  u[4] = cz * m10r - sz * m10i; u[5] = cz * m10i + sz * m10r;
  u[6] = cz * m11r - sz * m11i; u[7] = cz * m11i + sz * m11r;
}

__device__ __forceinline__ void async_gl2lds_b32(const void* g, void* l) {
  unsigned lds = (unsigned)(unsigned long long)(uintptr_t)l;  // low 32 bits = LDS offset
  unsigned long long ga = (unsigned long long)(uintptr_t)g;
  asm volatile("global_load_async_to_lds_b32 %0, %1, off" :: "v"(lds), "v"(ga) : "memory");
}

__global__ __launch_bounds__(BLOCK) void qsim_kernel(const float* __restrict__ x,
                                                     const float* __restrict__ w,
                                                     float* __restrict__ out) {
  __shared__ float  s_x[IN_DIM];
  __shared__ float  s_hp[NPAIR];
  __shared__ float  s_w[NWTS];
  __shared__ float  s_gr[256];
  __shared__ float  s_gi[256];
  __shared__ float2 s_xch[BLOCK * LOCAL];
  __shared__ float  s_red[8];

  const int t    = threadIdx.x;
  const int lane = t & 31;
  const int b    = blockIdx.x;

  // ---- async-stage x row and weights into LDS (CDNA5 async global->LDS path)
  if (t < IN_DIM)               async_gl2lds_b32(x + b * IN_DIM + t, &s_x[t]);
  else if (t < IN_DIM + NWTS)   async_gl2lds_b32(w + (t - IN_DIM),   &s_w[t - IN_DIM]);
  asm volatile("s_wait_asynccnt 0" ::: "memory");
  __syncthreads();
  if (t < NPAIR) s_hp[t] = 0.5f * s_x[NQ + t];
  __syncthreads();

  float re[LOCAL], im[LOCAL];

  // ---- initial state: H^12 then all x-dependent diagonal phases, folded analytically.
  // qubit i <-> bit (11-i) of flat index.
#pragma unroll 1
  for (int k = 0; k < LOCAL; ++k) {
    const unsigned idx = ((unsigned)t << 4) | (unsigned)k;
    float ang = 0.f;
#pragma unroll
    for (int i = 0; i < NQ; ++i)
      ang += ((idx >> (11 - i)) & 1u) ? s_x[i] : 0.f;
    int c = 0;
#pragma unroll
    for (int i = 0; i < NQ; ++i) {
#pragma unroll
      for (int j = i + 1; j < NQ; ++j) {
        const unsigned sg = ((idx >> (11 - i)) ^ (idx >> (11 - j))) & 1u;
        const float h = s_hp[c++];
        ang += sg ? h : -h;   // -0.5*theta*z_i*z_j
      }
    }
    float sv, cv;
    sincosf(ang, &sv, &cv);
    re[k] = cv * (1.f / 64.f);
    im[k] = sv * (1.f / 64.f);
  }

  // ---- 3 layers
#pragma unroll 1
  for (int l = 0; l < 3; ++l) {
    const int wb = l * 36;

    // Build G = U8 (x) U9 (x) U10 (x) U11 (16x16 complex) into LDS: one entry per thread.
    __syncthreads();
    {
      const int r = t >> 4, cc = t & 15;
      float gr = 1.f, gi = 0.f;
#pragma unroll
      for (int qq = 0; qq < 4; ++qq) {
        float u[8];
        const int q = 8 + qq;
        fused_u(s_w[wb + q * 3], s_w[wb + q * 3 + 1], s_w[wb + q * 3 + 2], u);
        const int rb = (r >> (3 - qq)) & 1, cb = (cc >> (3 - qq)) & 1;
        const float ur = rb ? (cb ? u[6] : u[4]) : (cb ? u[2] : u[0]);
        const float ui = rb ? (cb ? u[7] : u[5]) : (cb ? u[3] : u[1]);
        const float nr = gr * ur - gi * ui;
        gi = gr * ui + gi * ur;
        gr = nr;
      }
      s_gr[t] = gr;
      s_gi[t] = gi;
    }
    __syncthreads();

    // Apply G to local qubits: C = G * S as two 16x16x16 f32 matmuls per wave
    // (columns = lanes 0..15 -> group A, lanes 16..31 -> group B), each split into
    // 4 chained V_WMMA_F32_16X16X4_F32 per real-arithmetic combination.
    {
      const bool lo = lane < 16;
      float shr[LOCAL], shi[LOCAL];
#pragma unroll
      for (int j = 0; j < LOCAL; ++j) {
        shr[j] = __shfl_xor(re[j], 16);
        shi[j] = __shfl_xor(im[j], 16);
      }
      v8f crA = {0.f,0.f,0.f,0.f,0.f,0.f,0.f,0.f};
      v8f ciA = {0.f,0.f,0.f,0.f,0.f,0.f,0.f,0.f};
      v8f crB = {0.f,0.f,0.f,0.f,0.f,0.f,0.f,0.f};
      v8f ciB = {0.f,0.f,0.f,0.f,0.f,0.f,0.f,0.f};
      const int row   = lane & 15;
      const int khalf = lane >> 4;
#pragma unroll
      for (int kk = 0; kk < 4; ++kk) {
        // A operand (G), 16x4 f32 layout: lane = M + 16*(K/2), vgpr = K%2
        const int colb = 4 * kk + 2 * khalf;
        const float2 g2r = *reinterpret_cast<const float2*>(&s_gr[row * 16 + colb]);
        const float2 g2i = *reinterpret_cast<const float2*>(&s_gi[row * 16 + colb]);
        const v2f ar = { g2r.x, g2r.y };
        const v2f ai = { g2i.x, g2i.y };
        // B operand (state columns), 4x16 layout: lane = N + 16*(K/2), vgpr = K%2
        const v2f brA = { lo ? re[4*kk]   : shr[4*kk+2], lo ? re[4*kk+1] : shr[4*kk+3] };
        const v2f biA = { lo ? im[4*kk]   : shi[4*kk+2], lo ? im[4*kk+1] : shi[4*kk+3] };
        const v2f brB = { lo ? shr[4*kk]  : re[4*kk+2],  lo ? shr[4*kk+1]: re[4*kk+3] };
        const v2f biB = { lo ? shi[4*kk]  : im[4*kk+2],  lo ? shi[4*kk+1]: im[4*kk+3] };
        const v2f biAn = -biA, biBn = -biB;
        crA = __builtin_amdgcn_wmma_f32_16x16x4_f32(false, ar, false, brA,  (short)0, crA, false, false);
        crA = __builtin_amdgcn_wmma_f32_16x16x4_f32(false, ai, false, biAn, (short)0, crA, false, false);
        ciA = __builtin_amdgcn_wmma_f32_16x16x4_f32(false, ar, false, biA,  (short)0, ciA, false, false);
        ciA = __builtin_amdgcn_wmma_f32_16x16x4_f32(false, ai, false, brA,  (short)0, ciA, false, false);
        crB = __builtin_amdgcn_wmma_f32_16x16x4_f32(false, ar, false, brB,  (short)0, crB, false, false);
        crB = __builtin_amdgcn_wmma_f32_16x16x4_f32(false, ai, false, biBn, (short)0, crB, false, false);
        ciB = __builtin_amdgcn_wmma_f32_16x16x4_f32(false, ar, false, biB,  (short)0, ciB, false, false);
        ciB = __builtin_amdgcn_wmma_f32_16x16x4_f32(false, ai, false, brB,  (short)0, ciB, false, false);
      }
      // Unmarshal D (16x16 f32: vgpr r: lanes<16 -> M=r,N=lane ; lanes>=16 -> M=r+8,N=lane-16)
#pragma unroll
      for (int r2 = 0; r2 < 8; ++r2) {
        const float tAr = __shfl_xor(crA[r2], 16), tBr = __shfl_xor(crB[r2], 16);
        const float tAi = __shfl_xor(ciA[r2], 16), tBi = __shfl_xor(ciB[r2], 16);
        re[r2]     = lo ? crA[r2] : tBr;
        re[r2 + 8] = lo ? tAr     : crB[r2];
        im[r2]     = lo ? ciA[r2] : tBi;
        im[r2 + 8] = lo ? tAi     : ciB[r2];
      }
    }

    // Lane-bit qubits 3..7 (partner within wave via shfl_xor)
#pragma unroll 1
    for (int q = 3; q <= 7; ++q) {
      float u[8];
      fused_u(s_w[wb + q * 3], s_w[wb + q * 3 + 1], s_w[wb + q * 3 + 2], u);
      const int bs = 7 - q;
      const int m  = 1 << bs;
      const unsigned bit = ((unsigned)lane >> bs) & 1u;
      const float ar = bit ? u[6] : u[0], ai = bit ? u[7] : u[1];
      const float br = bit ? u[4] : u[2], bi = bit ? u[5] : u[3];
#pragma unroll
      for (int k = 0; k < LOCAL; ++k) {
        const float pr = __shfl_xor(re[k], m);
        const float pi = __shfl_xor(im[k], m);
        const float nr = ar * re[k] - ai * im[k] + br * pr - bi * pi;
        const float ni = ar * im[k] + ai * re[k] + br * pi + bi * pr;
        re[k] = nr; im[k] = ni;
      }
    }

    // Wave-bit qubits 0..2 (partner thread via LDS exchange)
#pragma unroll 1
    for (int q = 0; q <= 2; ++q) {
      float u[8];
      fused_u(s_w[wb + q * 3], s_w[wb + q * 3 + 1], s_w[wb + q * 3 + 2], u);
      const int bs = 7 - q;
      const int partner = t ^ (1 << bs);
      const unsigned bit = ((unsigned)t >> bs) & 1u;
      const float ar = bit ? u[6] : u[0], ai = bit ? u[7] : u[1];
      const float br = bit ? u[4] : u[2], bi = bit ? u[5] : u[3];
#pragma unroll
      for (int k = 0; k < LOCAL; ++k) s_xch[t * LOCAL + k] = make_float2(re[k], im[k]);
      __syncthreads();
#pragma unroll
      for (int k = 0; k < LOCAL; ++k) {
        const float2 p = s_xch[partner * LOCAL + k];
        const float nr = ar * re[k] - ai * im[k] + br * p.x - bi * p.y;
        const float ni = ar * im[k] + ai * re[k] + br * p.y + bi * p.x;
        re[k] = nr; im[k] = ni;
      }
      __syncthreads();
    }

    // CZ ring (layers 0,1): sign = parity over 11 adjacent-bit pairs + wrap(bit11,bit0)
    if (l < 2) {
#pragma unroll
      for (int k = 0; k < LOCAL; ++k) {
        const unsigned idx = ((unsigned)t << 4) | (unsigned)k;
        const unsigned par = ((unsigned)__popc(idx & (idx >> 1) & 0x7FFu) +
                             ((idx >> 11) & idx & 1u)) & 1u;
        re[k] = par ? -re[k] : re[k];
        im[k] = par ? -im[k] : im[k];
      }
    }
  }

  // ---- <Z0>: qubit0 = bit 11 of idx = bit 7 of t
  float acc = 0.f;
#pragma unroll
  for (int k = 0; k < LOCAL; ++k) acc += re[k] * re[k] + im[k] * im[k];
  if (t & 128) acc = -acc;
#pragma unroll
  for (int off = 16; off; off >>= 1) acc += __shfl_xor(acc, off);
  if (lane == 0) s_red[t >> 5] = acc;
  __syncthreads();
  if (t == 0) {
    float s = 0.f;
#pragma unroll
    for (int i = 0; i < 8; ++i) s += s_red[i];
    out[b] = s;
  }
}

extern "C" void kernel_launch(void* const* d_in, const int* in_sizes, int n_in,
                              void* d_out, int out_size, void* d_ws, size_t ws_size,
                              hipStream_t stream) {
  (void)in_sizes; (void)n_in; (void)out_size; (void)d_ws; (void)ws_size;
  const float* x = (const float*)d_in[0];
  const float* w = (const float*)d_in[1];
  float* out = (float*)d_out;
  qsim_kernel<<<256, BLOCK, 0, stream>>>(x, w, out);
}